// EdmondsSampler_20899310862745
// MI455X (gfx1250) — compile-verified
//
#include <hip/hip_runtime.h>
#include <hip/hip_bf16.h>
#include <math.h>

// MI455X / gfx1250, wave32. One 256-thread block (8 waves) per batch element
// for the Edmonds phase; z stays L2-resident (33MB << 192MB L2) so the
// iterative column rescans never touch HBM after the first pass.

#define NMAX 256
#define NEG_INF (-3.0e38f)

typedef float v2f __attribute__((ext_vector_type(2)));
typedef float v8f __attribute__((ext_vector_type(8)));

__global__ void __launch_bounds__(256)
gumbel_z_kernel(const float* __restrict__ logits, const float* __restrict__ u,
                float* __restrict__ arb, float* __restrict__ stats,
                float* __restrict__ zout, int total, int B) {
    int idx = blockIdx.x * blockDim.x + threadIdx.x;
    if (idx < total) {
        const float EPSF = 1.1920928955078125e-07f;     // np.finfo(f32).eps
        const float ONE_M = 0.99999988079071044921875f; // 1 - eps in f32
        float uu = u[idx];
        uu = fminf(fmaxf(uu, EPSF), ONE_M);
        // z = logits + (-log(-log(uc)))
        zout[idx] = logits[idx] - logf(-logf(uu));
        arb[idx] = 0.0f;
    }
    if (idx < B) stats[idx] = 0.0f;
}

__global__ void __launch_bounds__(256)
edmonds_kernel(const float* __restrict__ z, const int* __restrict__ lengths,
               float* __restrict__ arb, float* __restrict__ stats, int N) {
    __shared__ unsigned short comp[NMAX];      // current supernode label per node
    __shared__ unsigned short parentS[NMAX];   // union-find forest over supernode ids
    __shared__ unsigned short linktime[NMAX];  // contraction index when id was merged
    __shared__ unsigned short colsrc[NMAX];    // per-column best source (original row)
    __shared__ float          colval[NMAX];    // per-column best raw score
    __shared__ float          offv[NMAX];      // per-column accumulated reduction
    __shared__ float          sup_val[NMAX];   // per-supernode best reduced in-edge
    __shared__ unsigned short sup_i[NMAX], sup_j[NMAX], sup_head[NMAX];
    __shared__ unsigned short enter_src[NMAX], enter_dst[NMAX];
    __shared__ unsigned char  activeS[NMAX], incyc[NMAX], colorA[NMAX];
    __shared__ unsigned short path[NMAX], cyc[NMAX];
    __shared__ unsigned short log_m[2 * NMAX], log_i[2 * NMAX], log_j[2 * NMAX];
    __shared__ unsigned short con_w[NMAX], con_start[NMAX], con_len[NMAX];
    __shared__ float          sval[NMAX];
    __shared__ int s_ncon, s_logpos, s_done;

    const int b = blockIdx.x;
    const int tid = threadIdx.x;
    int L = lengths[b];
    if (L < 1) L = 1;
    if (L > N) L = N;
    const float* zb = z + (size_t)b * N * N;

    // ---- init ----
    comp[tid]     = (unsigned short)tid;
    parentS[tid]  = 0xFFFFu;
    linktime[tid] = 0xFFFFu;
    activeS[tid]  = (tid < L) ? 1 : 0;
    incyc[tid]    = 0;
    offv[tid]     = 0.0f;
    enter_src[tid] = 0;
    enter_dst[tid] = (unsigned short)tid;
    if (tid == 0) { s_ncon = 0; s_logpos = 0; s_done = 0; }
    __syncthreads();

    // ---- initial per-column argmax (exclude diagonal; column 0 forbidden) ----
    {
        float best = NEG_INF; int bi = 0;
        if (tid >= 1 && tid < L) {
            for (int i = 0; i < L; ++i) {
                if (i == tid) continue;
                float v = zb[(size_t)i * N + tid];
                if (v > best) { best = v; bi = i; }
            }
        }
        colval[tid] = best;
        colsrc[tid] = (unsigned short)bi;
    }
    __syncthreads();

    // ---- contraction loop ----
    for (int iter = 0; iter < NMAX + 2; ++iter) {
        // per-supernode best entering edge (reduced scores)
        {
            int v = tid;
            if (v >= 1 && v < L && activeS[v]) {
                float best = NEG_INF; int bj = 0;
                for (int j = 1; j < L; ++j) {
                    if (comp[j] == (unsigned short)v) {
                        float val = colval[j] - offv[j];
                        if (val > best) { best = val; bj = j; }
                    }
                }
                sup_val[v]  = best;
                sup_j[v]    = (unsigned short)bj;
                sup_i[v]    = colsrc[bj];
                sup_head[v] = comp[colsrc[bj]];
            }
        }
        __syncthreads();

        if (tid == 0) {
            // cycle detection over supernode head graph (root id 0 excluded)
            for (int v = 0; v < L; ++v) colorA[v] = 0;
            colorA[0] = 2;
            int cyclen = 0;
            for (int start = 1; start < L && cyclen == 0; ++start) {
                if (!activeS[start] || colorA[start]) continue;
                int plen = 0, v = start;
                while (colorA[v] == 0) {
                    colorA[v] = 1; path[plen++] = (unsigned short)v;
                    v = sup_head[v];
                }
                if (colorA[v] == 1) {
                    int i0 = 0;
                    while (path[i0] != (unsigned short)v) ++i0;
                    for (int k = i0; k < plen; ++k) cyc[cyclen++] = path[k];
                }
                for (int k = 0; k < plen; ++k) colorA[path[k]] = 2;
            }
            if (cyclen == 0) {
                s_done = 1;
            } else {
                int t = s_ncon;
                int W = cyc[0];                 // representative keeps its id
                con_w[t] = (unsigned short)W;
                con_start[t] = (unsigned short)s_logpos;
                con_len[t] = (unsigned short)cyclen;
                for (int k = 0; k < cyclen; ++k) {
                    int m = cyc[k];
                    log_m[s_logpos + k] = (unsigned short)m;
                    log_i[s_logpos + k] = sup_i[m];
                    log_j[s_logpos + k] = sup_j[m];
                    incyc[m] = 1;
                    if (m != W) {
                        parentS[m]  = (unsigned short)W;
                        linktime[m] = (unsigned short)t;
                        activeS[m]  = 0;
                    }
                }
                s_logpos += cyclen;
                s_ncon = t + 1;
            }
        }
        __syncthreads();
        if (s_done) break;

        const int W = con_w[s_ncon - 1];
        // apply reductions + relabel merged nodes
        if (tid < L) {
            int c = comp[tid];
            if (incyc[c]) {
                offv[tid] += sup_val[c];
                comp[tid] = (unsigned short)W;
            }
        }
        __syncthreads();
        incyc[tid] = 0;  // clear for next round
        // recompute column maxes only for columns inside the merged supernode
        if (tid >= 1 && tid < L && comp[tid] == (unsigned short)W) {
            float best = NEG_INF; int bi = 0;
            for (int i = 0; i < L; ++i) {
                if (comp[i] == (unsigned short)W) continue;
                float v = zb[(size_t)i * N + tid];
                if (v > best) { best = v; bi = i; }
            }
            colval[tid] = best;
            colsrc[tid] = (unsigned short)bi;
        }
        __syncthreads();
    }

    // ---- termination: record entering edges for surviving supernodes ----
    if (tid >= 1 && tid < L && activeS[tid]) {
        enter_src[tid] = sup_i[tid];
        enter_dst[tid] = sup_j[tid];
    }
    __syncthreads();

    // ---- replay contractions in reverse to resolve original heads ----
    if (tid == 0) {
        for (int t = s_ncon - 1; t >= 0; --t) {
            int W  = con_w[t];
            int es = enter_src[W];
            int ed = enter_dst[W];
            // which cycle member held column `ed` just before contraction t?
            int v = ed;
            while (parentS[v] != 0xFFFFu && (int)linktime[v] < t) v = parentS[v];
            int member = (parentS[v] != 0xFFFFu && (int)linktime[v] == t) ? v : W;
            int st = con_start[t], ln = con_len[t];
            for (int k = 0; k < ln; ++k) {
                int m = log_m[st + k];
                if (m == member) {
                    enter_src[m] = (unsigned short)es;
                    enter_dst[m] = (unsigned short)ed;
                } else {
                    enter_src[m] = log_i[st + k];
                    enter_dst[m] = log_j[st + k];
                }
            }
        }
    }
    __syncthreads();

    // ---- write arb, gather selected z values ----
    {
        float sv = 0.0f;
        if (tid >= 1 && tid < L) {
            int hd = enter_src[tid];
            arb[(size_t)b * N * N + (size_t)hd * N + tid] = 1.0f;
            sv = zb[(size_t)hd * N + tid];
        }
        sval[tid] = sv;
    }
    __syncthreads();

    // ---- stats[b] = sum(sval) via v_wmma_f32_16x16x4_f32 row-sum trick ----
    // View sval as a 16x16 matrix M; D = sum_k (M[:,4k:4k+4] x ones(4x16))
    // puts row-sums of M in every column of the f32 accumulator.
    if (tid < 32) {
        const int m  = tid & 15;   // A-matrix row for this lane
        const int hi = tid >> 4;   // lane group selects K pair {2hi, 2hi+1}
        v8f acc = {};
        v2f bmat; bmat.x = 1.0f; bmat.y = 1.0f;
#pragma unroll
        for (int kc = 0; kc < 4; ++kc) {
            v2f a;
            a.x = sval[16 * m + 4 * kc + 2 * hi];
            a.y = sval[16 * m + 4 * kc + 2 * hi + 1];
            acc = __builtin_amdgcn_wmma_f32_16x16x4_f32(
                false, a, false, bmat, (short)0, acc, false, false);
        }
        // lane 0 holds row-sums M=0..7 (N=0), lane 16 holds M=8..15 (N=0)
        float s = acc[0] + acc[1] + acc[2] + acc[3] +
                  acc[4] + acc[5] + acc[6] + acc[7];
        float tot = __shfl(s, 0, 32) + __shfl(s, 16, 32);
        if (tid == 0) stats[b] = tot;
    }
}

extern "C" void kernel_launch(void* const* d_in, const int* in_sizes, int n_in,
                              void* d_out, int out_size, void* d_ws, size_t ws_size,
                              hipStream_t stream) {
    const float* logits  = (const float*)d_in[0];
    const float* u       = (const float*)d_in[1];
    const int*   lengths = (const int*)d_in[2];

    const int B = in_sizes[2];              // 128
    int NN = in_sizes[0] / (B > 0 ? B : 1); // N*N
    int N = 1;
    while ((N + 1) * (N + 1) <= NN) ++N;    // integer sqrt
    if (N > NMAX) N = NMAX;                 // LDS arrays sized for N<=256

    const int total = B * N * N;
    float* arb   = (float*)d_out;
    float* statp = arb + (size_t)total;
    float* zout  = statp + B;

    gumbel_z_kernel<<<(total + 255) / 256, 256, 0, stream>>>(
        logits, u, arb, statp, zout, total, B);
    edmonds_kernel<<<B, 256, 0, stream>>>(zout, lengths, arb, statp, N);
}